// ROIPooler_52716428591915
// MI455X (gfx1250) — compile-verified
//
#include <hip/hip_runtime.h>

// FPN ROIPooler (ROIAlign, aligned=True) for MI455X / gfx1250.
// TDM-staged version: per box, the Tensor Data Mover DMAs the box's feature
// region (rw1 x rh1 x 8 channels, row stride W, channel stride H*W) into LDS
// (double buffered), and all bilinear gathers run out of LDS. Each DRAM byte
// of the region is read exactly once per box; VMEM only does coalesced stores.

#define OUT_HW  14
#define NBINS   196            // 14*14
#define NSAMP   28             // 14 bins * sampling ratio 2
#define NCH     256
#define NBOXF   5
#define GCH     8              // channels per TDM descriptor
#define NGRP    (NCH / GCH)    // 32 groups
#define MAXTILE 1152           // floats per channel tile; worst case ~1060

typedef unsigned int        u32;
typedef unsigned long long  u64;
typedef u32 u32x4 __attribute__((ext_vector_type(4)));
typedef int i32x4 __attribute__((ext_vector_type(4)));
typedef int i32x8 __attribute__((ext_vector_type(8)));

// torchvision ROIAlign axis weights (edge handling identical to reference)
__device__ __forceinline__ void axis_tab(float coord, float L,
                                         int& lo, int& hi, float& wl, float& wh) {
    const bool valid = (coord >= -1.0f) && (coord <= L);
    float c   = fminf(fmaxf(coord, 0.0f), L - 1.0f);
    float lof = floorf(c);
    float hif = fminf(lof + 1.0f, L - 1.0f);
    wh = c - lof;
    wl = 1.0f - wh;
    if (!valid) { wl = 0.0f; wh = 0.0f; }
    lo = (int)lof;
    hi = (int)hif;
}

__device__ __forceinline__ void tdm_issue(u32 lds_addr, u64 gaddr,
                                          i32x8 g1, i32x4 g2, i32x4 g3) {
    u32x4 g0;
    g0[0] = 1u;                                 // count=1, is_restore=0, gather off
    g0[1] = lds_addr;                           // lds_addr (bytes) at [63:32]
    g0[2] = (u32)gaddr;                         // global_addr [95:64]
    g0[3] = (u32)(gaddr >> 32) | (2u << 30);    // global_addr hi + type=2 ("image")
#if __has_builtin(__builtin_amdgcn_tensor_load_to_lds)
#if defined(__clang_major__) && (__clang_major__ >= 23)
    i32x8 gz = (i32x8)(0);
    __builtin_amdgcn_tensor_load_to_lds(g0, g1, g2, g3, gz, 0);
#else
    __builtin_amdgcn_tensor_load_to_lds(g0, g1, g2, g3, 0);
#endif
#endif
}

__global__ __launch_bounds__(256) void roipooler_tdm_gfx1250_kernel(
    const float* __restrict__ x0, const float* __restrict__ x1,
    const float* __restrict__ x2, const float* __restrict__ x3,
    const float* __restrict__ boxes, float* __restrict__ out)
{
    // Packed tables: .x = bitcast(lo), .y = bitcast(hi), .z = w_lo, .w = w_hi
    __shared__ float4 s_y[NSAMP];
    __shared__ float4 s_x[NSAMP];
    __shared__ float  s_tile[2][GCH * MAXTILE];   // double-buffered TDM landing zone

    const int m   = blockIdx.x;
    const int tid = threadIdx.x;

    // ---- uniform per-box scalars (scalarized; compiler uses s_*_f32 pipe)
    const float bidxf = boxes[NBOXF * m + 0];
    const float bx0   = boxes[NBOXF * m + 1];
    const float by0   = boxes[NBOXF * m + 2];
    const float bx1   = boxes[NBOXF * m + 3];
    const float by1   = boxes[NBOXF * m + 4];
    const int   bidx  = (int)bidxf;

    // level: floor(4 + log2(sqrt(w*h)/224 + 1e-8)), clip [2,5], -2
    const float size = sqrtf((bx1 - bx0) * (by1 - by0));
    float lvlf = floorf(4.0f + log2f(size * (1.0f / 224.0f) + 1e-8f));
    lvlf = fminf(fmaxf(lvlf, 2.0f), 5.0f);
    const int lvl = (int)lvlf - 2;                    // 0..3

    const float scale = 0.25f / (float)(1 << lvl);    // exact powers of two
    const int   Hdim  = 256 >> lvl;                   // square maps
    const float* __restrict__ base =
        (lvl == 0) ? x0 : (lvl == 1) ? x1 : (lvl == 2) ? x2 : x3;

    const float rx0  = bx0 * scale - 0.5f;            // aligned=True
    const float ry0  = by0 * scale - 0.5f;
    const float binW = ((bx1 * scale - 0.5f) - rx0) * (1.0f / (float)OUT_HW);
    const float binH = ((by1 * scale - 0.5f) - ry0) * (1.0f / (float)OUT_HW);

    // ---- bilinear tables (28 y-samples, 28 x-samples) into LDS
    if (tid < NSAMP) {
        const int k = tid;
        const float s = (float)(k >> 1) + ((float)(k & 1) + 0.5f) * 0.5f;
        int lo, hi; float wl, wh;
        axis_tab(ry0 + s * binH, (float)Hdim, lo, hi, wl, wh);
        s_y[k] = make_float4(__int_as_float(lo), __int_as_float(hi), wl, wh);
    } else if (tid >= 32 && tid < 32 + NSAMP) {
        const int k = tid - 32;
        const float s = (float)(k >> 1) + ((float)(k & 1) + 0.5f) * 0.5f;
        int lo, hi; float wl, wh;
        axis_tab(rx0 + s * binW, (float)Hdim, lo, hi, wl, wh);
        s_x[k] = make_float4(__int_as_float(lo), __int_as_float(hi), wl, wh);
    }
    __syncthreads();

    // ---- box region bounds (tables are monotonic in the sample index)
    const int ymin = __float_as_int(s_y[0].x);
    const int ymax = __float_as_int(s_y[NSAMP - 1].y);
    const int xmin = __float_as_int(s_x[0].x);
    const int xmax = __float_as_int(s_x[NSAMP - 1].y);
    const int rw1  = xmax - xmin + 1;                  // <= ~130
    const int rh1  = ymax - ymin + 1;
    int tileA = rw1 * rh1;                             // <= ~1060 by level bound
    if (tileA > MAXTILE) tileA = MAXTILE;              // defensive, never expected

    const int chanStride = Hdim * Hdim;
    const float* __restrict__ fb =
        base + (size_t)bidx * NCH * (size_t)chanStride + (size_t)ymin * Hdim + xmin;
    float* __restrict__ ob = out + (size_t)m * (NCH * NBINS);

    // ---- D# invariants (CDNA5 ISA): 3D tile rw1 x rh1 x GCH
    i32x8 g1;
    g1[0] = 0x00020000;                                        // data_size=2 (4B)
    g1[1] = (int)((rw1 & 0xFFFF) << 16);                       // tensor_dim0 lo16
    g1[2] = (int)(((u32)rw1 >> 16) | ((u32)(rh1 & 0xFFFF) << 16)); // td0 hi | td1 lo
    g1[3] = (int)(((u32)rh1 >> 16) | ((u32)(rw1 & 0xFFFF) << 16)); // td1 hi | tile_dim0
    g1[4] = (int)((u32)(rh1 & 0xFFFF) | ((u32)GCH << 16));     // tile_dim1 | tile_dim2
    g1[5] = Hdim;                                              // tensor_dim0_stride (row)
    g1[6] = (int)(((u32)chanStride & 0xFFFF) << 16);           // t1_stride lo16 (channel)
    g1[7] = (int)((u32)chanStride >> 16);                      // t1_stride hi
    i32x4 g2; g2[0] = GCH; g2[1] = 0; g2[2] = 0; g2[3] = 0;    // tensor_dim2
    i32x4 g3; g3[0] = 0;   g3[1] = 0; g3[2] = 0; g3[3] = 0;

    // LDS byte offsets: low 32 bits of a generic LDS pointer are the LDS address
    const u32 ldsB[2] = { (u32)(uintptr_t)(const void*)&s_tile[0][0],
                          (u32)(uintptr_t)(const void*)&s_tile[1][0] };
    const bool isW0 = (tid < 32);                     // one wave drives the TDM

    if (isW0) {
        tdm_issue(ldsB[0], (u64)(uintptr_t)(const void*)fb, g1, g2, g3);
    }

    for (int grp = 0; grp < NGRP; ++grp) {
        const int par = grp & 1;
        if (isW0) {
            if (grp + 1 < NGRP) {
                const float* gp = fb + (size_t)((grp + 1) * GCH) * chanStride;
                tdm_issue(ldsB[par ^ 1], (u64)(uintptr_t)(const void*)gp, g1, g2, g3);
#if __has_builtin(__builtin_amdgcn_s_wait_tensorcnt)
                __builtin_amdgcn_s_wait_tensorcnt(1);   // current group landed
            } else {
                __builtin_amdgcn_s_wait_tensorcnt(0);
#endif
            }
        }
        __syncthreads();                                // tile[par] visible to all waves

        const float* __restrict__ tile = &s_tile[par][0];
        const int c0 = grp * GCH;

        for (int e = tid; e < GCH * NBINS; e += 256) {
            const int cc  = e / NBINS;
            const int bin = e - cc * NBINS;
            const int ph  = bin / OUT_HW;
            const int pw  = bin - ph * OUT_HW;

            const float4 ty0 = s_y[2 * ph + 0];
            const float4 ty1 = s_y[2 * ph + 1];
            const float4 tx0 = s_x[2 * pw + 0];
            const float4 tx1 = s_x[2 * pw + 1];

            const int r0 = (__float_as_int(ty0.x) - ymin) * rw1;  // sample0 row lo
            const int r1 = (__float_as_int(ty0.y) - ymin) * rw1;  // sample0 row hi
            const int r2 = (__float_as_int(ty1.x) - ymin) * rw1;  // sample1 row lo
            const int r3 = (__float_as_int(ty1.y) - ymin) * rw1;  // sample1 row hi
            const int c0x = __float_as_int(tx0.x) - xmin;
            const int c1x = __float_as_int(tx0.y) - xmin;
            const int c2x = __float_as_int(tx1.x) - xmin;
            const int c3x = __float_as_int(tx1.y) - xmin;

            const float* __restrict__ tp = tile + cc * tileA;

            float acc;
            acc  = ty0.z * (tx0.z * tp[r0 + c0x] + tx0.w * tp[r0 + c1x])
                 + ty0.w * (tx0.z * tp[r1 + c0x] + tx0.w * tp[r1 + c1x]);   // (y0,x0)
            acc += ty0.z * (tx1.z * tp[r0 + c2x] + tx1.w * tp[r0 + c3x])
                 + ty0.w * (tx1.z * tp[r1 + c2x] + tx1.w * tp[r1 + c3x]);   // (y0,x1)
            acc += ty1.z * (tx0.z * tp[r2 + c0x] + tx0.w * tp[r2 + c1x])
                 + ty1.w * (tx0.z * tp[r3 + c0x] + tx0.w * tp[r3 + c1x]);   // (y1,x0)
            acc += ty1.z * (tx1.z * tp[r2 + c2x] + tx1.w * tp[r2 + c3x])
                 + ty1.w * (tx1.z * tp[r3 + c2x] + tx1.w * tp[r3 + c3x]);   // (y1,x1)

            ob[(size_t)(c0 + cc) * NBINS + bin] = acc * 0.25f;  // mean of 2x2 samples
        }
        __syncthreads();          // all reads of tile[par] done before its reuse
    }
}

extern "C" void kernel_launch(void* const* d_in, const int* in_sizes, int n_in,
                              void* d_out, int out_size, void* d_ws, size_t ws_size,
                              hipStream_t stream) {
    const float* x0    = (const float*)d_in[0];   // [2,256,256,256]
    const float* x1    = (const float*)d_in[1];   // [2,256,128,128]
    const float* x2    = (const float*)d_in[2];   // [2,256, 64, 64]
    const float* x3    = (const float*)d_in[3];   // [2,256, 32, 32]
    const float* boxes = (const float*)d_in[4];   // [512,5]
    float* out = (float*)d_out;                   // [512,256,14,14]

    const int M = in_sizes[4] / NBOXF;            // 512 boxes
    roipooler_tdm_gfx1250_kernel<<<dim3(M), dim3(256), 0, stream>>>(
        x0, x1, x2, x3, boxes, out);
}